// GCNLayer_39771397161472
// MI455X (gfx1250) — compile-verified
//
#include <hip/hip_runtime.h>

typedef __attribute__((ext_vector_type(2))) float v2f;
typedef __attribute__((ext_vector_type(8))) float v8f;

#define D_FEAT 64

// ---------------------------------------------------------------------------
// Kernel 1: X_norm = X * norm ; agg = X_norm  (self-loop init)
// float4-vectorized; 16 float4 per row of 64 features.
// ---------------------------------------------------------------------------
__global__ void gcn_init_kernel(const float4* __restrict__ X,
                                const float* __restrict__ norm,
                                float4* __restrict__ xnorm,
                                float4* __restrict__ agg,
                                int total4) {
    int i = blockIdx.x * blockDim.x + threadIdx.x;
    if (i >= total4) return;
    float nv = norm[i >> 4];          // 16 float4 per row
    float4 x = X[i];
    float4 v;
    v.x = x.x * nv; v.y = x.y * nv; v.z = x.z * nv; v.w = x.w * nv;
    xnorm[i] = v;
    agg[i]   = v;
}

// ---------------------------------------------------------------------------
// Kernel 2: undirected edge scatter.
// One edge per wave32 (ra/rb wave-uniform), float2 per lane -> covers 64 feats.
// agg[rb] += xnorm[ra]; agg[ra] += xnorm[rb]
// Relaxed agent-scope fetch_add with discarded result -> hardware
// GLOBAL_ATOMIC_ADD_F32 in no-return form (STOREcnt path), NOT a CAS loop.
// xnorm/agg both fit in the 192MB L2, so this runs at L2 atomic throughput.
// ---------------------------------------------------------------------------
__device__ __forceinline__ void atomic_add_f32_agent(float* p, float v) {
    (void)__hip_atomic_fetch_add(p, v, __ATOMIC_RELAXED, __HIP_MEMORY_SCOPE_AGENT);
}

__global__ void gcn_scatter_kernel(const float* __restrict__ xnorm,
                                   float* __restrict__ agg,
                                   const int* __restrict__ ref_a,
                                   const int* __restrict__ ref_b,
                                   int n_edges) {
    long tid  = (long)blockIdx.x * blockDim.x + threadIdx.x;
    int  edge = (int)(tid >> 5);
    if (edge >= n_edges) return;
    int  f  = ((int)tid & 31) * 2;
    long ra = ref_a[edge];
    long rb = ref_b[edge];
    float2 va = *(const float2*)(xnorm + ra * D_FEAT + f);
    float2 vb = *(const float2*)(xnorm + rb * D_FEAT + f);
    float* da = agg + rb * D_FEAT + f;
    float* db = agg + ra * D_FEAT + f;
    atomic_add_f32_agent(da,     va.x);
    atomic_add_f32_agent(da + 1, va.y);
    atomic_add_f32_agent(db,     vb.x);
    atomic_add_f32_agent(db + 1, vb.y);
}

// ---------------------------------------------------------------------------
// Kernel 3: out = relu((agg * norm) @ W + b) using V_WMMA_F32_16X16X4_F32.
// One wave computes a 16-row x 64-col output tile:
//   4 N-tiles x 16 K-steps = 64 WMMA ops per wave.
// A fragment (16x4 f32): lanes 0-15 carry K=0,1; lanes 16-31 carry K=2,3
//   for row M = m0 + (lane & 15). norm rescale folded into the A load.
// B fragment (4x16 f32): v0 = {K=0 | K=2} striped over lane halves, v1 = {K=1 | K=3}.
// D (16x16 f32): VGPR i -> rows m0+i (lanes 0-15) / m0+i+8 (lanes 16-31).
// ---------------------------------------------------------------------------
__global__ void __launch_bounds__(128)
gcn_gemm_kernel(const float* __restrict__ agg,
                const float* __restrict__ norm,
                const float* __restrict__ W,
                const float* __restrict__ bias,
                float* __restrict__ out,
                int n_rows) {
    int tile = blockIdx.x * (blockDim.x >> 5) + ((int)threadIdx.x >> 5);
    int m0   = tile * 16;
    if (m0 >= n_rows) return;              // whole-wave exit; EXEC stays all-1s

    int lane  = (int)threadIdx.x & 31;
    int mlo   = lane & 15;
    int mrow  = m0 + mlo;
    int khalf = (lane >> 4) << 1;          // 0 for lanes 0-15, 2 for lanes 16-31

    float nrm = norm[mrow];
    const float* arow = agg + (size_t)mrow * D_FEAT;

    v8f c0 = {}, c1 = {}, c2 = {}, c3 = {};

#pragma unroll
    for (int k = 0; k < 16; ++k) {
        int kc = k * 4 + khalf;
        // A fragment: (agg * norm) for K = kc, kc+1 of row mrow
        float2 av = *(const float2*)(arow + kc);
        v2f a;
        a[0] = av.x * nrm;
        a[1] = av.y * nrm;
        // B fragments for the four 16-column tiles of W (row-major 64x64)
        const float* wp = W + (size_t)kc * D_FEAT + mlo;
        v2f b0; b0[0] = wp[0];  b0[1] = wp[D_FEAT + 0];
        v2f b1; b1[0] = wp[16]; b1[1] = wp[D_FEAT + 16];
        v2f b2; b2[0] = wp[32]; b2[1] = wp[D_FEAT + 32];
        v2f b3; b3[0] = wp[48]; b3[1] = wp[D_FEAT + 48];

        c0 = __builtin_amdgcn_wmma_f32_16x16x4_f32(false, a, false, b0, (short)0, c0, false, false);
        c1 = __builtin_amdgcn_wmma_f32_16x16x4_f32(false, a, false, b1, (short)0, c1, false, false);
        c2 = __builtin_amdgcn_wmma_f32_16x16x4_f32(false, a, false, b2, (short)0, c2, false, false);
        c3 = __builtin_amdgcn_wmma_f32_16x16x4_f32(false, a, false, b3, (short)0, c3, false, false);
    }

    // Epilogue: bias + relu, D layout: VGPR i -> row m0+i (lanes 0-15) / +8 (hi)
    int rbase = m0 + ((lane >> 4) << 3);
    float bv0 = bias[mlo];
    float bv1 = bias[16 + mlo];
    float bv2 = bias[32 + mlo];
    float bv3 = bias[48 + mlo];

#pragma unroll
    for (int i = 0; i < 8; ++i) {
        size_t o = (size_t)(rbase + i) * D_FEAT + mlo;
        out[o]      = fmaxf(c0[i] + bv0, 0.0f);
        out[o + 16] = fmaxf(c1[i] + bv1, 0.0f);
        out[o + 32] = fmaxf(c2[i] + bv2, 0.0f);
        out[o + 48] = fmaxf(c3[i] + bv3, 0.0f);
    }
}

// ---------------------------------------------------------------------------
// Launch
// ---------------------------------------------------------------------------
extern "C" void kernel_launch(void* const* d_in, const int* in_sizes, int n_in,
                              void* d_out, int out_size, void* d_ws, size_t ws_size,
                              hipStream_t stream) {
    const float* X     = (const float*)d_in[0];   // [N, 64]
    const float* norm  = (const float*)d_in[1];   // [N, 1]
    const int*   ref_a = (const int*)d_in[2];     // [E]
    const int*   ref_b = (const int*)d_in[3];     // [E]
    const float* W     = (const float*)d_in[4];   // [64, 64]
    const float* bias  = (const float*)d_in[5];   // [64]
    float*       out   = (float*)d_out;           // [N, 64]

    const int N = in_sizes[1];                    // norm is [N,1]
    const int E = in_sizes[2];

    float* xnorm = (float*)d_ws;                       // N*64 f32
    float* agg   = xnorm + (size_t)N * D_FEAT;         // N*64 f32

    // Phase 1: X_norm and agg init
    int total4 = (N * D_FEAT) / 4;
    gcn_init_kernel<<<(total4 + 255) / 256, 256, 0, stream>>>(
        (const float4*)X, norm, (float4*)xnorm, (float4*)agg, total4);

    // Phase 2: edge scatter (one edge per wave32)
    long sthreads = (long)E * 32;
    int  sblocks  = (int)((sthreads + 255) / 256);
    gcn_scatter_kernel<<<sblocks, 256, 0, stream>>>(xnorm, agg, ref_a, ref_b, E);

    // Phase 3: WMMA projection + bias + relu
    int tiles  = (N + 15) / 16;                   // 6250
    int wpb    = 128 / 32;                        // 4 waves per block
    int blocks = (tiles + wpb - 1) / wpb;
    gcn_gemm_kernel<<<blocks, 128, 0, stream>>>(agg, norm, W, bias, out, N);
}